// Attention_89154931130951
// MI455X (gfx1250) — compile-verified
//
#include <hip/hip_runtime.h>
#include <math.h>
#include <stdint.h>

// Problem constants (from reference setup): B=4, L=1024, D=512, U=64.
#define B_DIM 4
#define L_DIM 1024
#define D_DIM 512
#define U_DIM 64
#define EPS_F 1e-7f

#define QK_STRIDE 68   // 16B-aligned rows, banks (4*row+u)%64 -> conflict-free
#define E_STRIDE  18   // 8B-aligned WMMA-A pairs, banks (18*r+c)%64 distinct

typedef __attribute__((ext_vector_type(2))) float v2f;
typedef __attribute__((ext_vector_type(8))) float v8f;

// D(16x16,f32) = A(16x4,f32) x B(4x16,f32) + C   -- V_WMMA_F32_16X16X4_F32
__device__ __forceinline__ v8f wmma4(v2f a, v2f b, v8f c) {
    return __builtin_amdgcn_wmma_f32_16x16x4_f32(
        /*neg_a=*/false, a, /*neg_b=*/false, b,
        /*c_mod=*/(short)0, c, /*reuse_a=*/false, /*reuse_b=*/false);
}

// Native gfx1250 V_TANH_F32 (trans op). Builtin if available (compiler manages
// the trans-op co-execution hazard); else inline asm with a trailing v_nop per
// ISA 7.4.3 (a trans result needs one independent op or V_NOP before it is used).
__device__ __forceinline__ float fast_tanh(float x) {
#if __has_builtin(__builtin_amdgcn_tanhf)
    return __builtin_amdgcn_tanhf(x);
#else
    float r;
    asm volatile("v_tanh_f32 %0, %1\n\tv_nop" : "=v"(r) : "v"(x));
    return r;
#endif
}

// Async global->LDS 16-byte copy (GLOBAL_LOAD_ASYNC_TO_LDS_B128, ASYNCcnt).
// lds_off is the 32-bit LDS byte offset (low 32 bits of the generic pointer:
// the shared aperture occupies addr[63:32] only, ISA §10.2).
__device__ __forceinline__ void async_copy_b128(uint32_t lds_off, const float* g) {
    asm volatile("global_load_async_to_lds_b128 %0, %1, off"
                 :: "v"(lds_off), "v"(g) : "memory");
}
__device__ __forceinline__ void wait_async0() {
    asm volatile("s_wait_asynccnt 0x0" ::: "memory");
}

// ---------------------------------------------------------------------------
// Kernel 1: qws = x @ Wt + bt (bias folded in), kws = x @ Wx
// grid = 256 (16-row M tiles over B*L), block = 128 (4 waves; wave w owns
// N-tile w of U=64 for BOTH q and k).
// ---------------------------------------------------------------------------
__global__ __launch_bounds__(128) void proj_kernel(
        const float* __restrict__ x,
        const float* __restrict__ Wx,
        const float* __restrict__ Wt,
        const float* __restrict__ bt,
        float* __restrict__ qws,
        float* __restrict__ kws) {
    const int m0   = blockIdx.x * 16;
    const int wave = threadIdx.x >> 5;
    const int lane = threadIdx.x & 31;
    const int half = lane >> 4;      // 0: K0/K1 side, 1: K2/K3 side
    const int ml   = lane & 15;      // A-matrix row within tile
    const int n    = wave * 16 + ml; // B/C column (== u index)

    const float* xrow = x + (size_t)(m0 + ml) * D_DIM;
    const float  btn  = bt[n];

    v8f cq, ck;
    for (int r = 0; r < 8; ++r) { cq[r] = 0.0f; ck[r] = 0.0f; }

    for (int k0 = 0; k0 < D_DIM; k0 += 4) {
        const int kb = k0 + 2 * half;
        // A fragment: lane<16 -> {K0,K1}, lane>=16 -> {K2,K3} of row ml
        v2f a;  a[0]  = xrow[kb];               a[1]  = xrow[kb + 1];
        // B fragments: VGPR0 holds row kb, VGPR1 holds row kb+1, col n
        v2f bq; bq[0] = Wt[kb * U_DIM + n];     bq[1] = Wt[(kb + 1) * U_DIM + n];
        v2f bk; bk[0] = Wx[kb * U_DIM + n];     bk[1] = Wx[(kb + 1) * U_DIM + n];
        cq = wmma4(a, bq, cq);
        ck = wmma4(a, bk, ck);
    }

    for (int r = 0; r < 8; ++r) {
        const int row = m0 + r + 8 * half;     // C/D layout: VGPR r <-> row r+8*half
        qws[row * U_DIM + n] = cq[r] + btn;    // fold bt into q
        kws[row * U_DIM + n] = ck[r];
    }
}

// ---------------------------------------------------------------------------
// Kernel 2: fused score + normalize + a@x, with async double-buffered k tiles.
//   v_i = (sum_j e_ij * x_j) / (sum_j e_ij + eps),
//   e_ij = exp( sum_u Wa_u * tanh(qb_iu + k_ju) + ba ),  qb = q + bt
// grid = 256 (one 16-row i tile each), block = 256 (8 waves; wave w owns
// a 64-wide slice of D=512 in 4 WMMA N-tiles).
// ---------------------------------------------------------------------------
__global__ __launch_bounds__(256) void attn_kernel(
        const float* __restrict__ x,
        const float* __restrict__ qws,
        const float* __restrict__ kws,
        const float* __restrict__ Wa,
        const float* __restrict__ ba,
        float* __restrict__ out) {
    __shared__ __align__(16) float sh_q[16 * QK_STRIDE];
    __shared__ __align__(16) float sh_kbuf[2][16 * QK_STRIDE];  // double buffer
    __shared__ __align__(16) float sh_wa[U_DIM];
    __shared__ __align__(8)  float sh_e[16 * E_STRIDE];
    __shared__ float red[256];
    __shared__ float sh_s[16];

    const int t    = threadIdx.x;
    const int row0 = blockIdx.x * 16;          // global row in [0, B*L)
    const int b    = row0 / L_DIM;             // 1024 % 16 == 0, tile never spans batches
    const int wave = t >> 5;
    const int lane = t & 31;
    const int half = lane >> 4;
    const int nl   = lane & 15;
    const int n0   = wave * 64;                // this wave's D-slice base

    const size_t xbase = (size_t)b * L_DIM * D_DIM;
    const size_t kbase = (size_t)b * L_DIM * U_DIM;

    // Per-thread slot of a 16x64 k tile: one b128 each (256 * 16B = 4KB tile)
    const int kr = t >> 4;                     // tile row
    const int kc = (t & 15) * 4;               // starting column
    const float*   kgsrc = kws + kbase + (size_t)kr * U_DIM + kc;  // + j0*U_DIM per tile
    const uint32_t lds_k0 = (uint32_t)(uintptr_t)&sh_kbuf[0][kr * QK_STRIDE + kc];
    const uint32_t lds_k1 = (uint32_t)(uintptr_t)&sh_kbuf[1][kr * QK_STRIDE + kc];

    // Kick off async copy of k tile 0 into buffer 0
    async_copy_b128(lds_k0, kgsrc);

    // Stage q tile [16 x 64] (bias already folded) and Wa
    for (int idx = t; idx < 16 * U_DIM; idx += 256) {
        const int r = idx >> 6, c = idx & 63;
        sh_q[r * QK_STRIDE + c] = qws[(row0 + r) * U_DIM + c];
    }
    if (t < U_DIM) sh_wa[t] = Wa[t];
    const float bav = ba[0];

    const int ei = t >> 4;                     // this thread's score row
    const int ej = t & 15;                     // this thread's score col
    float my_sum = 0.0f;

    v8f acc[4];
    for (int q = 0; q < 4; ++q)
        for (int r = 0; r < 8; ++r) acc[q][r] = 0.0f;

    for (int j0 = 0; j0 < L_DIM; j0 += 16) {
        const int cur = (j0 >> 4) & 1;
        // Publish current k tile: my async write done, then everyone's.
        // (The barrier also means all threads finished reading buf[1-cur].)
        wait_async0();
        __syncthreads();

        // Prefetch next k tile into the other buffer; overlaps score + WMMA.
        if (j0 + 16 < L_DIM)
            async_copy_b128(cur ? lds_k0 : lds_k1,
                            kgsrc + (size_t)(j0 + 16) * U_DIM);

        // --- score: one e_ij per thread; native V_TANH_F32, float4 LDS reads ---
        {
            const float4* qv = (const float4*)(&sh_q[ei * QK_STRIDE]);
            const float4* kv = (const float4*)(&sh_kbuf[cur][ej * QK_STRIDE]);
            const float4* wv = (const float4*)(sh_wa);
            float s = 0.0f;
            #pragma unroll
            for (int u4 = 0; u4 < U_DIM / 4; ++u4) {
                const float4 q4 = qv[u4];
                const float4 k4 = kv[u4];
                const float4 w4 = wv[u4];
                s += w4.x * fast_tanh(q4.x + k4.x);
                s += w4.y * fast_tanh(q4.y + k4.y);
                s += w4.z * fast_tanh(q4.z + k4.z);
                s += w4.w * fast_tanh(q4.w + k4.w);
            }
            const float e = expf(s + bav);
            sh_e[ei * E_STRIDE + ej] = e;
            my_sum += e;
        }
        __syncthreads();

        // --- V(16x64 slice) += E(16x16) @ X_j(16x64 slice) via WMMA f32 ---
        for (int kk = 0; kk < 4; ++kk) {
            const int kb = kk * 4 + 2 * half;
            v2f a;
            a[0] = sh_e[nl * E_STRIDE + kb];
            a[1] = sh_e[nl * E_STRIDE + kb + 1];
            const float* xr0 = x + xbase + (size_t)(j0 + kb) * D_DIM;
            const float* xr1 = xr0 + D_DIM;
            #pragma unroll
            for (int tt = 0; tt < 4; ++tt) {
                const int n = n0 + tt * 16 + nl;
                v2f bx; bx[0] = xr0[n]; bx[1] = xr1[n];
                acc[tt] = wmma4(a, bx, acc[tt]);
            }
        }
    }

    // --- reduce row sums of e over the 16 j-columns held per thread ---
    red[t] = my_sum;
    __syncthreads();
    if (t < 16) {
        float s = 0.0f;
        for (int c = 0; c < 16; ++c) s += red[t * 16 + c];
        sh_s[t] = s + EPS_F;
    }
    __syncthreads();

    // --- normalize and store ---
    for (int tt = 0; tt < 4; ++tt) {
        #pragma unroll
        for (int r = 0; r < 8; ++r) {
            const int row = r + 8 * half;      // C/D layout
            const float val = acc[tt][r] / sh_s[row];
            out[(size_t)(row0 + row) * D_DIM + n0 + tt * 16 + nl] = val;
        }
    }
}

// ---------------------------------------------------------------------------
extern "C" void kernel_launch(void* const* d_in, const int* in_sizes, int n_in,
                              void* d_out, int out_size, void* d_ws, size_t ws_size,
                              hipStream_t stream) {
    (void)in_sizes; (void)n_in; (void)out_size; (void)ws_size;
    const float* x  = (const float*)d_in[0];  // [B,L,D]
    const float* Wx = (const float*)d_in[1];  // [D,U]
    const float* Wt = (const float*)d_in[2];  // [D,U]
    const float* bt = (const float*)d_in[3];  // [U]
    const float* Wa = (const float*)d_in[4];  // [U,1]
    const float* ba = (const float*)d_in[5];  // [1]
    float* out = (float*)d_out;               // [B,L,D]

    float* qws = (float*)d_ws;                      // [B*L, U] (q + bt)
    float* kws = qws + (size_t)B_DIM * L_DIM * U_DIM;

    const int mtiles = (B_DIM * L_DIM) / 16;  // 256
    proj_kernel<<<mtiles, 128, 0, stream>>>(x, Wx, Wt, bt, qws, kws);
    attn_kernel<<<mtiles, 256, 0, stream>>>(x, qws, kws, Wa, ba, out);
}